// ScaleGNN_89421219102735
// MI455X (gfx1250) — compile-verified
//
#include <hip/hip_runtime.h>
#include <hip/hip_bf16.h>
#include <math.h>

// ---------------------------------------------------------------------------
// ScaleGNN for MI455X (gfx1250, wave32).
// All dense GEMMs use V_WMMA_F32_16X16X4_F32 (faithful fp32; the problem is
// bandwidth-bound at 23.3 TB/s, so no precision downgrade is warranted).
// A^2@x is computed as A@(A@x) to avoid the 4096^3 A@A matmul.
// The random mask reproduces JAX threefry2x32 exactly (key = {0,42}).
// Each wave computes a 16x64 tile with 4 independent WMMA accumulators so
// consecutive WMMAs have no RAW dependence and the matrix pipe stays busy.
// ---------------------------------------------------------------------------

#define NN 4096
#define CC 128
#define HH 128
#define OO 64

typedef __attribute__((ext_vector_type(2))) float v2f;
typedef __attribute__((ext_vector_type(8))) float v8f;

// ---------------------------------------------------------------------------
// Generic WMMA f32 GEMM: D = A(M x K) * B(K x N)   (optionally B given as
// row-major N x K and used transposed).  One wave per 16x64 output tile
// (4 accumulators), 4 waves (64 output rows) per block.
// K multiple of 16; M multiple of 64; N multiple of 64.
// EPI: 0 = none, 1 = +bias[col], 2 = relu(+bias[col]), 3 = *scale[row]
// ---------------------------------------------------------------------------
template <bool BT, int EPI>
__global__ __launch_bounds__(128) void gemm_wmma_f32(
    const float* __restrict__ A, const float* __restrict__ B,
    const float* __restrict__ aux, float* __restrict__ D,
    int K, int lda, int ldb, int ldd) {
  const int lane = threadIdx.x & 31;
  const int wave = threadIdx.x >> 5;
  const int half = lane >> 4;    // 0: K pair {0,1}/rows M..M+7, 1: {2,3}/M+8..
  const int lrow = lane & 15;
  const int m0 = (blockIdx.y * 4 + wave) * 16;
  const int n0 = blockIdx.x * 64;

  const v8f vzero = {0.f, 0.f, 0.f, 0.f, 0.f, 0.f, 0.f, 0.f};
  v8f acc0 = vzero, acc1 = vzero, acc2 = vzero, acc3 = vzero;

  // A fragment: lanes 0-15 hold K={k,k+1} of row m0+lrow,
  //             lanes 16-31 hold K={k+2,k+3}            (ISA 7.12.2)
  const float* arow = A + (size_t)(m0 + lrow) * lda + 2 * half;
  // BT path: four row-panels of B^T, 16 cols apart
  const float* br0 = B + (size_t)(n0 + lrow) * ldb + 2 * half;
  const float* br1 = br0 + (size_t)16 * ldb;
  const float* br2 = br1 + (size_t)16 * ldb;
  const float* br3 = br2 + (size_t)16 * ldb;

  for (int k = 0; k < K; k += 16) {
    __builtin_prefetch(arow + k + 256, 0, 3);   // global_prefetch_b8
#pragma unroll
    for (int kk = 0; kk < 16; kk += 4) {
      const int kb = k + kk;
      v2f a = *(const v2f*)(arow + kb);
      v2f b0, b1, b2, b3;
      if (BT) {
        b0 = *(const v2f*)(br0 + kb);
        b1 = *(const v2f*)(br1 + kb);
        b2 = *(const v2f*)(br2 + kb);
        b3 = *(const v2f*)(br3 + kb);
      } else {
        const int kr = kb + 2 * half;
        const float* Bp = B + (size_t)kr * ldb + n0 + lrow;
        b0.x = Bp[0];  b0.y = Bp[ldb];
        b1.x = Bp[16]; b1.y = Bp[16 + ldb];
        b2.x = Bp[32]; b2.y = Bp[32 + ldb];
        b3.x = Bp[48]; b3.y = Bp[48 + ldb];
      }
      // 4 independent accumulators: no WMMA->WMMA RAW, shared A fragment.
      acc0 = __builtin_amdgcn_wmma_f32_16x16x4_f32(false, a, false, b0,
                                                   (short)0, acc0, false, false);
      acc1 = __builtin_amdgcn_wmma_f32_16x16x4_f32(false, a, false, b1,
                                                   (short)0, acc1, false, false);
      acc2 = __builtin_amdgcn_wmma_f32_16x16x4_f32(false, a, false, b2,
                                                   (short)0, acc2, false, false);
      acc3 = __builtin_amdgcn_wmma_f32_16x16x4_f32(false, a, false, b3,
                                                   (short)0, acc3, false, false);
    }
  }

  // C/D layout: VGPR v -> row m0 + v + 8*half, col = tile_n0 + (lane&15)
  const int c0 = n0 + lrow;
  float bias0 = 0.f, bias1 = 0.f, bias2 = 0.f, bias3 = 0.f;
  if (EPI == 1 || EPI == 2) {
    bias0 = aux[c0]; bias1 = aux[c0 + 16];
    bias2 = aux[c0 + 32]; bias3 = aux[c0 + 48];
  }
#pragma unroll
  for (int v = 0; v < 8; ++v) {
    const int row = m0 + v + 8 * half;
    float* Drow = D + (size_t)row * ldd;
    float v0 = acc0[v], v1 = acc1[v], v2 = acc2[v], v3 = acc3[v];
    if (EPI == 1) { v0 += bias0; v1 += bias1; v2 += bias2; v3 += bias3; }
    if (EPI == 2) {
      v0 = fmaxf(v0 + bias0, 0.f); v1 = fmaxf(v1 + bias1, 0.f);
      v2 = fmaxf(v2 + bias2, 0.f); v3 = fmaxf(v3 + bias3, 0.f);
    }
    if (EPI == 3) {
      const float s = aux[row];
      v0 *= s; v1 *= s; v2 *= s; v3 *= s;
    }
    Drow[c0] = v0; Drow[c0 + 16] = v1; Drow[c0 + 32] = v2; Drow[c0 + 48] = v3;
  }
}

// ---------------------------------------------------------------------------
// Small helper kernels
// ---------------------------------------------------------------------------
__global__ void zero_f32(float* __restrict__ p, unsigned n) {
  unsigned i = blockIdx.x * blockDim.x + threadIdx.x;
  if (i < n) p[i] = 0.f;
}

// A[src, dst] = 1.0 (idempotent store -> duplicate edges collapse like .set)
__global__ void scatter_edges(const long long* __restrict__ ei,
                              float* __restrict__ A, int E) {
  int e = blockIdx.x * blockDim.x + threadIdx.x;
  if (e < E) {
    long long s = ei[e];
    long long d = ei[(size_t)E + e];
    A[(size_t)s * NN + d] = 1.0f;
  }
}

// JAX threefry2x32 mask: mask[idx] = (uniform(key(42))[idx] < 0.5)
// uniform<0.5  <=>  top bit of the 32 random bits is 0.
__device__ __forceinline__ unsigned rotl32(unsigned x, int d) {
  return (x << d) | (x >> (32 - d));
}
__global__ void gen_mask(float* __restrict__ Mf, unsigned halfN) {
  unsigned t = blockIdx.x * blockDim.x + threadIdx.x;
  if (t >= halfN) return;
  const unsigned k0 = 0u, k1 = 42u, k2 = k0 ^ k1 ^ 0x1BD11BDAu;
  unsigned x0 = t + k0;
  unsigned x1 = (t + halfN) + k1;
#define TFR(r) { x0 += x1; x1 = rotl32(x1, r); x1 ^= x0; }
  TFR(13) TFR(15) TFR(26) TFR(6)   x0 += k1; x1 += k2 + 1u;
  TFR(17) TFR(29) TFR(16) TFR(24)  x0 += k2; x1 += k0 + 2u;
  TFR(13) TFR(15) TFR(26) TFR(6)   x0 += k0; x1 += k1 + 3u;
  TFR(17) TFR(29) TFR(16) TFR(24)  x0 += k1; x1 += k2 + 4u;
  TFR(13) TFR(15) TFR(26) TFR(6)   x0 += k2; x1 += k0 + 5u;
#undef TFR
  Mf[t]          = (x0 < 0x80000000u) ? 1.0f : 0.0f;
  Mf[t + halfN]  = (x1 < 0x80000000u) ? 1.0f : 0.0f;
}

// row-normalize x -> xn = x / (||x|| + 1e-8).  one wave per row (C=128)
__global__ void normalize_rows(const float* __restrict__ x,
                               float* __restrict__ xn) {
  const int row = blockIdx.x * 8 + (threadIdx.x >> 5);
  const int lane = threadIdx.x & 31;
  float v[4], s = 0.f;
#pragma unroll
  for (int i = 0; i < 4; ++i) {
    v[i] = x[(size_t)row * CC + lane + 32 * i];
    s += v[i] * v[i];
  }
#pragma unroll
  for (int off = 16; off > 0; off >>= 1) s += __shfl_xor(s, off, 32);
  const float inv = 1.f / (sqrtf(s) + 1e-8f);
#pragma unroll
  for (int i = 0; i < 4; ++i)
    xn[(size_t)row * CC + lane + 32 * i] = v[i] * inv;
}

// per-row top-10 of sim (ties -> lowest index, matching lax.top_k);
// force Mf[row, idx] = 1.
__global__ __launch_bounds__(256) void topk_force(
    const float* __restrict__ sim, float* __restrict__ Mf) {
  __shared__ float sv[NN];
  __shared__ float rv[256];
  __shared__ int ri[256];
  const int row = blockIdx.x;
  const int tid = threadIdx.x;
  for (int j = tid; j < NN; j += 256) sv[j] = sim[(size_t)row * NN + j];
  __syncthreads();
  for (int t = 0; t < 10; ++t) {
    float bv = -INFINITY; int bi = NN;
    const int base = tid * (NN / 256);
#pragma unroll 4
    for (int j = 0; j < NN / 256; ++j) {
      float v = sv[base + j];
      if (v > bv) { bv = v; bi = base + j; }
    }
    rv[tid] = bv; ri[tid] = bi;
    __syncthreads();
    for (int s = 128; s > 0; s >>= 1) {
      if (tid < s) {
        float ov = rv[tid + s]; int oi = ri[tid + s];
        if (ov > rv[tid] || (ov == rv[tid] && oi < ri[tid])) {
          rv[tid] = ov; ri[tid] = oi;
        }
      }
      __syncthreads();
    }
    if (tid == 0) {
      Mf[(size_t)row * NN + ri[0]] = 1.0f;
      sv[ri[0]] = -INFINITY;
    }
    __syncthreads();
  }
}

// agg = a0*x + a1*y1 + a2*y2
__global__ void combine_hops(const float* __restrict__ x,
                             const float* __restrict__ y1,
                             const float* __restrict__ y2,
                             const float* __restrict__ alpha,
                             float* __restrict__ agg, unsigned n) {
  unsigned i = blockIdx.x * blockDim.x + threadIdx.x;
  if (i < n) agg[i] = alpha[0] * x[i] + alpha[1] * y1[i] + alpha[2] * y2[i];
}

// invdeg[row] = 1 / (sum_j Mf[row,j] + 1e-6).  one wave per row.
__global__ void inv_rowsum(const float* __restrict__ Mf,
                           float* __restrict__ invdeg) {
  const int row = blockIdx.x * 8 + (threadIdx.x >> 5);
  const int lane = threadIdx.x & 31;
  float s = 0.f;
  for (int j = lane; j < NN; j += 32) s += Mf[(size_t)row * NN + j];
#pragma unroll
  for (int off = 16; off > 0; off >>= 1) s += __shfl_xor(s, off, 32);
  if (lane == 0) invdeg[row] = 1.f / (s + 1e-6f);
}

// xf = concat([x, fused], axis=-1)  (4096 x 256)
__global__ void concat_cols(const float* __restrict__ x,
                            const float* __restrict__ fus,
                            float* __restrict__ xf, unsigned n) {
  unsigned i = blockIdx.x * blockDim.x + threadIdx.x;
  if (i >= n) return;
  unsigned row = i >> 8, c = i & 255;
  xf[i] = (c < CC) ? x[(size_t)row * CC + c] : fus[(size_t)row * CC + (c - CC)];
}

// log_softmax over 64 cols; one wave per row, 2 elems/lane.
__global__ void log_softmax64(const float* __restrict__ z,
                              float* __restrict__ out) {
  const int row = blockIdx.x * 8 + (threadIdx.x >> 5);
  const int lane = threadIdx.x & 31;
  float a = z[(size_t)row * OO + lane];
  float b = z[(size_t)row * OO + lane + 32];
  float m = fmaxf(a, b);
#pragma unroll
  for (int off = 16; off > 0; off >>= 1) m = fmaxf(m, __shfl_xor(m, off, 32));
  float s = expf(a - m) + expf(b - m);
#pragma unroll
  for (int off = 16; off > 0; off >>= 1) s += __shfl_xor(s, off, 32);
  const float lse = m + logf(s);
  out[(size_t)row * OO + lane] = a - lse;
  out[(size_t)row * OO + lane + 32] = b - lse;
}

// ---------------------------------------------------------------------------
extern "C" void kernel_launch(void* const* d_in, const int* in_sizes, int n_in,
                              void* d_out, int out_size, void* d_ws,
                              size_t ws_size, hipStream_t stream) {
  const float* x        = (const float*)d_in[0];
  const long long* ei   = (const long long*)d_in[1];
  const float* alpha    = (const float*)d_in[2];
  const float* fusion_W = (const float*)d_in[3];
  const float* fusion_b = (const float*)d_in[4];
  const float* lin0_W   = (const float*)d_in[5];
  const float* lin0_b   = (const float*)d_in[6];
  const float* lin1_W   = (const float*)d_in[7];
  const float* lin1_b   = (const float*)d_in[8];
  const int E = in_sizes[1] / 2;

  // workspace layout (floats)
  float* W = (float*)d_ws;
  const size_t NNsq = (size_t)NN * NN;       // 16,777,216
  float* Abuf = W;
  float* Mf   = Abuf + NNsq;
  float* sim  = Mf + NNsq;
  float* y1   = sim + NNsq;                  // N x C
  float* y2   = y1 + (size_t)NN * CC;
  float* xn   = y2 + (size_t)NN * CC;
  float* agg  = xn + (size_t)NN * CC;
  float* fus  = agg + (size_t)NN * CC;
  float* xf   = fus + (size_t)NN * CC;       // N x 2H
  float* h0   = xf + (size_t)NN * 2 * HH;
  float* h1   = h0 + (size_t)NN * HH;
  float* zz   = h1 + (size_t)NN * HH;        // N x O
  float* invd = zz + (size_t)NN * OO;        // N

  const dim3 blk(256);

  // 1) dense adjacency
  zero_f32<<<dim3(NNsq / 256), blk, 0, stream>>>(Abuf, (unsigned)NNsq);
  scatter_edges<<<dim3((E + 255) / 256), blk, 0, stream>>>(ei, Abuf, E);

  // 2) threefry random keep-mask (full overwrite each call -> deterministic)
  gen_mask<<<dim3((unsigned)(NNsq / 2 / 256)), blk, 0, stream>>>(
      Mf, (unsigned)(NNsq / 2));

  // 3) xn, y1 = A@x, y2 = A@y1  (== A^2 @ x)
  normalize_rows<<<dim3(NN / 8), blk, 0, stream>>>(x, xn);
  gemm_wmma_f32<false, 0><<<dim3(CC / 64, NN / 64), dim3(128), 0, stream>>>(
      Abuf, x, nullptr, y1, NN, NN, CC, CC);
  gemm_wmma_f32<false, 0><<<dim3(CC / 64, NN / 64), dim3(128), 0, stream>>>(
      Abuf, y1, nullptr, y2, NN, NN, CC, CC);

  // 4) sim = xn @ xn^T, then top-10 forces mask bits
  gemm_wmma_f32<true, 0><<<dim3(NN / 64, NN / 64), dim3(128), 0, stream>>>(
      xn, xn, nullptr, sim, CC, CC, CC, NN);
  topk_force<<<dim3(NN), blk, 0, stream>>>(sim, Mf);

  // 5) agg, degrees, fused = (Mf @ agg) * invdeg
  combine_hops<<<dim3((unsigned)(NN * CC / 256)), blk, 0, stream>>>(
      x, y1, y2, alpha, agg, (unsigned)(NN * CC));
  inv_rowsum<<<dim3(NN / 8), blk, 0, stream>>>(Mf, invd);
  gemm_wmma_f32<false, 3><<<dim3(CC / 64, NN / 64), dim3(128), 0, stream>>>(
      Mf, agg, invd, fus, NN, NN, CC, CC);

  // 6) MLP: h0 = [x|fused]@fusion_W + b ; h1 = relu(h0@lin0_W + b) ;
  //    zz = h1@lin1_W + b ; out = log_softmax(zz)
  concat_cols<<<dim3((unsigned)(NN * 2 * HH / 256)), blk, 0, stream>>>(
      x, fus, xf, (unsigned)(NN * 2 * HH));
  gemm_wmma_f32<false, 1><<<dim3(HH / 64, NN / 64), dim3(128), 0, stream>>>(
      xf, fusion_W, fusion_b, h0, 2 * HH, 2 * HH, HH, HH);
  gemm_wmma_f32<false, 2><<<dim3(HH / 64, NN / 64), dim3(128), 0, stream>>>(
      h0, lin0_W, lin0_b, h1, HH, HH, HH, HH);
  gemm_wmma_f32<false, 1><<<dim3(OO / 64, NN / 64), dim3(128), 0, stream>>>(
      h1, lin1_W, lin1_b, zz, HH, HH, OO, OO);
  log_softmax64<<<dim3(NN / 8), blk, 0, stream>>>(zz, (float*)d_out);
}